// GCN_34110630265038
// MI455X (gfx1250) — compile-verified
//
#include <hip/hip_runtime.h>
#include <hip/hip_bf16.h>

typedef float v2f __attribute__((ext_vector_type(2)));
typedef float v8f __attribute__((ext_vector_type(8)));

#define H 64          // hidden/feature width (F == H == 64)
#define C_CLS 16
#define N_GRAPHS 64
#define BN_EPS 1e-5f

// ---------------------------------------------------------------------------
// Dense GEMM: out[N,64] = A[N,64] @ W[64,64] using V_WMMA_F32_16X16X4_F32.
// Block = 4 waves; wave w owns column tile w (n0 = 16w); block b owns row tile.
// A frag (16x4 f32): lane L holds A[m0 + L%16][k0 + 2*(L/16) + v], v in {0,1}.
// B frag (4x16 f32): lane L holds W[k0 + 2*(L/16) + v][n0 + L%16].
// C/D (16x16 f32, 8 VGPRs): vgpr r -> row m0 + r + 8*(L/16), col n0 + L%16.
// ---------------------------------------------------------------------------
__global__ void __launch_bounds__(128)
gcn_gemm64(const float* __restrict__ A, const float* __restrict__ W,
           float* __restrict__ out)
{
    const int lane = threadIdx.x & 31;
    const int wave = threadIdx.x >> 5;
    const int m0   = blockIdx.x << 4;
    const int n0   = wave << 4;
    const int half = lane >> 4;   // 0 or 1
    const int l16  = lane & 15;

    v8f c = {0.f, 0.f, 0.f, 0.f, 0.f, 0.f, 0.f, 0.f};

    const float* arow = A + (size_t)(m0 + l16) * H + 2 * half;
#pragma unroll
    for (int k0 = 0; k0 < H; k0 += 4) {
        v2f a = *(const v2f*)(arow + k0);
        v2f b;
        b.x = W[(k0 + 2 * half + 0) * H + n0 + l16];
        b.y = W[(k0 + 2 * half + 1) * H + n0 + l16];
        c = __builtin_amdgcn_wmma_f32_16x16x4_f32(
                /*neg_a=*/false, a, /*neg_b=*/false, b,
                /*c_mod=*/(short)0, c, /*reuse_a=*/false, /*reuse_b=*/false);
    }

#pragma unroll
    for (int r = 0; r < 8; ++r)
        out[(size_t)(m0 + r + 8 * half) * H + n0 + l16] = c[r];
}

// ---------------------------------------------------------------------------
// Degree / normalization
// ---------------------------------------------------------------------------
__global__ void deg_init(float* __restrict__ deg, int N) {
    int i = blockIdx.x * blockDim.x + threadIdx.x;
    if (i < N) deg[i] = 1.0f;                       // self loop
}

__global__ void deg_count(const int* __restrict__ dst, float* __restrict__ deg, int E) {
    int e = blockIdx.x * blockDim.x + threadIdx.x;
    if (e < E) atomicAdd(&deg[dst[e]], 1.0f);
}

__global__ void deg_to_dinv(float* __restrict__ deg, int N) {
    int i = blockIdx.x * blockDim.x + threadIdx.x;
    if (i < N) deg[i] = rsqrtf(deg[i]);
}

// ---------------------------------------------------------------------------
// agg[v][c] = xw[v][c] * dinv[v]^2 + bias[c]   (self-loop term + bias init)
// ---------------------------------------------------------------------------
__global__ void __launch_bounds__(256)
agg_init(const float* __restrict__ xw, const float* __restrict__ dinv,
         const float* __restrict__ bias, float* __restrict__ agg)
{
    int idx = blockIdx.x * 256 + threadIdx.x;       // over N*H
    int n = idx >> 6, c = idx & 63;
    float di = dinv[n];
    agg[idx] = xw[idx] * di * di + bias[c];
}

// ---------------------------------------------------------------------------
// One wave32 per edge: coalesced 256B row gather, 64 L2 f32 atomics scatter.
// ---------------------------------------------------------------------------
__global__ void __launch_bounds__(256)
edge_scatter(const float* __restrict__ xw, const int* __restrict__ src,
             const int* __restrict__ dst, const float* __restrict__ dinv,
             float* __restrict__ agg, int E)
{
    int e = blockIdx.x * 8 + (threadIdx.x >> 5);
    if (e >= E) return;
    int s = src[e], d = dst[e];
    float nrm = dinv[s] * dinv[d];
    int j = (threadIdx.x & 31) * 2;
    float2 v = *(const float2*)(xw + (size_t)s * H + j);
    atomicAdd(&agg[(size_t)d * H + j + 0], v.x * nrm);
    atomicAdd(&agg[(size_t)d * H + j + 1], v.y * nrm);
}

// ---------------------------------------------------------------------------
// BatchNorm helpers
// ---------------------------------------------------------------------------
__global__ void zero_f32(float* __restrict__ p, int n) {
    int i = blockIdx.x * blockDim.x + threadIdx.x;
    if (i < n) p[i] = 0.0f;
}

// optional ReLU in place + per-column sum / sumsq accumulation.
// grid = N/256 blocks of 256 threads; thread t: column t&63, 64 strided rows.
__global__ void __launch_bounds__(256)
relu_stats(float* __restrict__ h, float* __restrict__ stats, int do_relu)
{
    int c  = threadIdx.x & 63;
    int r0 = blockIdx.x * 256 + (threadIdx.x >> 6);
    float s = 0.f, sq = 0.f;
#pragma unroll 8
    for (int i = 0; i < 64; ++i) {
        size_t idx = (size_t)(r0 + i * 4) * H + c;
        float v = h[idx];
        if (do_relu) v = fmaxf(v, 0.f);
        h[idx] = v;
        s += v; sq += v * v;
    }
    atomicAdd(&stats[c], s);
    atomicAdd(&stats[64 + c], sq);
}

__global__ void bn_params(float* __restrict__ stats, float invN) {
    int c = threadIdx.x;                            // 64 threads
    float mean = stats[c] * invN;
    float var  = stats[64 + c] * invN - mean * mean; // biased var, matches ref
    stats[c]      = mean;
    stats[64 + c] = rsqrtf(var + BN_EPS);
}

__global__ void __launch_bounds__(256)
bn_apply(float* __restrict__ h, const float* __restrict__ stats)
{
    int idx = blockIdx.x * 256 + threadIdx.x;       // over N*H
    int c = idx & 63;
    h[idx] = (h[idx] - stats[c]) * stats[64 + c];
}

// ---------------------------------------------------------------------------
// Global mean pool (atomic segment sum) + classifier + softmax
// ---------------------------------------------------------------------------
__global__ void __launch_bounds__(256)
pool_sum(const float* __restrict__ h, const int* __restrict__ batch,
         float* __restrict__ pooled, float* __restrict__ cnt)
{
    int idx = blockIdx.x * 256 + threadIdx.x;       // over N*H
    int n = idx >> 6, c = idx & 63;
    int g = batch[n];
    atomicAdd(&pooled[g * H + c], h[idx]);
    if (c == 0) atomicAdd(&cnt[g], 1.0f);
}

__global__ void __launch_bounds__(1024)
classify_softmax(const float* __restrict__ pooled, const float* __restrict__ cnt,
                 const float* __restrict__ clfW, const float* __restrict__ clfb,
                 float* __restrict__ out)
{
    int t = threadIdx.x;                            // 1024 = 64 graphs * 16 classes
    int g = t >> 4, cls = t & 15;
    float inv = 1.0f / fmaxf(cnt[g], 1.0f);
    float acc = clfb[cls];
#pragma unroll 8
    for (int k = 0; k < H; ++k)
        acc += pooled[g * H + k] * inv * clfW[cls * H + k];
    // softmax across the 16-lane class group (wave32 holds 2 graphs)
    float m = acc;
    for (int o = 8; o >= 1; o >>= 1) m = fmaxf(m, __shfl_xor(m, o, 16));
    float e = __expf(acc - m);
    float ssum = e;
    for (int o = 8; o >= 1; o >>= 1) ssum += __shfl_xor(ssum, o, 16);
    out[g * C_CLS + cls] = e / ssum;
}

// ---------------------------------------------------------------------------
// Host orchestration
// ---------------------------------------------------------------------------
extern "C" void kernel_launch(void* const* d_in, const int* in_sizes, int n_in,
                              void* d_out, int out_size, void* d_ws, size_t ws_size,
                              hipStream_t stream)
{
    const float* x      = (const float*)d_in[0];
    const int*   edge   = (const int*)  d_in[1];    // [2, E]
    const int*   batch  = (const int*)  d_in[2];
    const float* Ws[3]  = {(const float*)d_in[3], (const float*)d_in[5], (const float*)d_in[7]};
    const float* bs[3]  = {(const float*)d_in[4], (const float*)d_in[6], (const float*)d_in[8]};
    const float* clfW   = (const float*)d_in[9];
    const float* clfb   = (const float*)d_in[10];

    const int N = in_sizes[0] / H;                  // 65536
    const int E = in_sizes[1] / 2;                  // 1048576
    const int* src = edge;
    const int* dst = edge + E;

    float* ws     = (float*)d_ws;
    float* buf0   = ws;                             // N*H  (agg / h)
    float* buf1   = buf0 + (size_t)N * H;           // N*H  (xw)
    float* dinv   = buf1 + (size_t)N * H;           // N
    float* stats  = dinv + N;                       // 128
    float* pooled = stats + 128;                    // 64*64
    float* cnt    = pooled + N_GRAPHS * H;          // 64

    const int NH_blocks = (N * H) / 256;

    // degree -> dinv (edges are layer-invariant: compute once per call)
    deg_init   <<<(N + 255) / 256, 256, 0, stream>>>(dinv, N);
    deg_count  <<<(E + 255) / 256, 256, 0, stream>>>(dst, dinv, E);
    deg_to_dinv<<<(N + 255) / 256, 256, 0, stream>>>(dinv, N);

    const float* hin = x;
    for (int l = 0; l < 3; ++l) {
        gcn_gemm64  <<<N / 16, 128, 0, stream>>>(hin, Ws[l], buf1);
        agg_init    <<<NH_blocks, 256, 0, stream>>>(buf1, dinv, bs[l], buf0);
        edge_scatter<<<(E + 7) / 8, 256, 0, stream>>>(buf1, src, dst, dinv, buf0, E);
        zero_f32    <<<1, 128, 0, stream>>>(stats, 128);
        relu_stats  <<<N / 256, 256, 0, stream>>>(buf0, stats, (l < 2) ? 1 : 0);
        bn_params   <<<1, 64, 0, stream>>>(stats, 1.0f / (float)N);
        bn_apply    <<<NH_blocks, 256, 0, stream>>>(buf0, stats);
        hin = buf0;
    }

    zero_f32<<<(N_GRAPHS * H + N_GRAPHS + 255) / 256, 256, 0, stream>>>(pooled, N_GRAPHS * H + N_GRAPHS);
    pool_sum<<<NH_blocks, 256, 0, stream>>>(buf0, batch, pooled, cnt);
    classify_softmax<<<1, 1024, 0, stream>>>(pooled, cnt, clfW, clfb, (float*)d_out);
}